// IndRNN_46325517254838
// MI455X (gfx1250) — compile-verified
//
#include <hip/hip_runtime.h>
#include <stdint.h>

// IndRNN fused kernel for MI455X (gfx1250, wave32).
// gi = x @ W_ih^T + bias via V_WMMA_F32_16X16X4_F32 (exact fp32 path),
// recurrence h = relu(gi_t + h*w_hh) fused per 16-timestep chunk.
// x tiles double-buffered in LDS via global_load_async_to_lds_b128 (ASYNCcnt).

typedef float v2f __attribute__((ext_vector_type(2)));
typedef float v8f __attribute__((ext_vector_type(8)));

#define B_ 64
#define T_ 2048
#define I_ 128
#define H_ 512

#define XPITCH 132            // padded LDS pitch (floats) for a 16x128 x-tile
#define XBUF   (16 * XPITCH)  // floats per x-tile buffer
#define GIPITCH 34            // padded LDS pitch (floats) for the 16x32 gi scratch

// Async copy 16B global -> LDS, tracked with ASYNCcnt (no VGPR transit).
__device__ __forceinline__ void stage_async16(const float* gsrc, uint32_t lds_off) {
    uint64_t ga = (uint64_t)(uintptr_t)gsrc;
    asm volatile("global_load_async_to_lds_b128 %0, %1, off"
                 :: "v"(lds_off), "v"(ga) : "memory");
}

__global__ __launch_bounds__(256) void indrnn_fused_kernel(
    const float* __restrict__ x,      // [B, T, I]
    const float* __restrict__ w_ih,   // [H, I]
    const float* __restrict__ w_hh,   // [H]
    const float* __restrict__ b_ih,   // [H]
    float* __restrict__ out)          // [B*T*H] outputs ++ [B*H] h_last
{
    __shared__ float lds_x[2 * XBUF];               // double-buffered x tiles
    __shared__ float lds_gi[8 * 16 * GIPITCH];      // per-wave gi scratch

    const int tid  = threadIdx.x;
    const int lane = tid & 31;
    const int wave = tid >> 5;
    const int b    = blockIdx.x;                    // batch index
    const int hw   = blockIdx.y * 256 + wave * 32;  // this wave's 32 h-columns

    // ---- Preload B operands (W^T slices) into registers, ISA B-layout ----
    // B is 4x16 (KxN): lanes 0-15 hold K={k0,k0+1}, lanes 16-31 hold K={k0+2,k0+3}.
    const int koff = (lane >> 4) << 1;              // 0 or 2
    const int colA = hw + (lane & 15);              // N-tile 0 column
    const int colB = colA + 16;                     // N-tile 1 column
    v2f wb0[32], wb1[32];
#pragma unroll
    for (int kk = 0; kk < 32; ++kk) {
        wb0[kk] = *(const v2f*)(w_ih + colA * I_ + kk * 4 + koff);
        wb1[kk] = *(const v2f*)(w_ih + colB * I_ + kk * 4 + koff);
    }

    const float whh  = w_hh[hw + lane];
    const float bias = b_ih[hw + lane];
    float h = 0.0f;

    const float* xb   = x + (size_t)b * T_ * I_;
    float*       gw   = lds_gi + wave * (16 * GIPITCH);
    const int    arow = lane & 15;                  // A-matrix row = t within chunk
    const int    crow = (lane < 16) ? 0 : 8;        // C-tile row base for this lane half
    const int    ccol = lane & 15;

    // Per-thread staging geometry: two 16B chunks per thread per 8KB tile.
    const int e0 = tid * 4;                         // element index in tile
    const int e1 = (tid + 256) * 4;
    const uint32_t l0 = (uint32_t)(uintptr_t)(lds_x + (e0 >> 7) * XPITCH + (e0 & 127));
    const uint32_t l1 = (uint32_t)(uintptr_t)(lds_x + (e1 >> 7) * XPITCH + (e1 & 127));

    // ---- Prologue: stage chunk 0 into buffer 0 ----
    stage_async16(xb + e0, l0);
    stage_async16(xb + e1, l1);

    int cur = 0;
    for (int t0 = 0; t0 < T_; t0 += 16) {
        // Stage chunk t0+16 into the other buffer; wait only for the current one.
        if (t0 + 16 < T_) {
            const float* g = xb + (size_t)(t0 + 16) * I_;
            uint32_t nb = (uint32_t)((cur ^ 1) * (XBUF * 4));
            stage_async16(g + e0, l0 + nb);
            stage_async16(g + e1, l1 + nb);
            // in-order completion: <=2 outstanding == the 2 just issued
            asm volatile("s_wait_asynccnt 0x2" ::: "memory");
        } else {
            asm volatile("s_wait_asynccnt 0x0" ::: "memory");
        }
        if (t0 + 32 < T_)                           // warm L2 two chunks ahead
            __builtin_prefetch(xb + (size_t)(t0 + 32) * I_ + tid * 8, 0, 0);
        __syncthreads();

        // ---- gi tile: two 16x16 accumulators, K=128 in 32 WMMA k-steps ----
        const float* xt = lds_x + cur * XBUF;
        v8f c0 = {}, c1 = {};
#pragma unroll
        for (int kk = 0; kk < 32; ++kk) {
            v2f a = *(const v2f*)(xt + arow * XPITCH + kk * 4 + koff);
            c0 = __builtin_amdgcn_wmma_f32_16x16x4_f32(
                    false, a, false, wb0[kk], (short)0, c0, false, false);
            c1 = __builtin_amdgcn_wmma_f32_16x16x4_f32(
                    false, a, false, wb1[kk], (short)0, c1, false, false);
        }

        // ---- Spill C tiles (rows v / v+8 per VGPR) to wave-private LDS ----
#pragma unroll
        for (int v = 0; v < 8; ++v) {
            gw[(crow + v) * GIPITCH + ccol]      = c0[v];
            gw[(crow + v) * GIPITCH + ccol + 16] = c1[v];
        }
        __syncthreads();  // gi visible; all waves done reading x buffer `cur`

        // ---- Sequential recurrence over the 16 timesteps of this chunk ----
        float* orow = out + ((size_t)b * T_ + t0) * H_ + hw + lane;
#pragma unroll
        for (int r = 0; r < 16; ++r) {
            float g = gw[r * GIPITCH + lane] + bias;
            h = fmaxf(0.0f, g + h * whh);
            __builtin_nontemporal_store(h, orow + (size_t)r * H_);  // NT: keep L2 for x
        }
        cur ^= 1;
        // no extra barrier: next async stage targets the buffer whose DS reads
        // all completed before the barrier above; gi scratch is wave-private.
    }

    // ---- h_last, appended after outputs ----
    out[(size_t)B_ * T_ * H_ + (size_t)b * H_ + hw + lane] = h;
}

extern "C" void kernel_launch(void* const* d_in, const int* in_sizes, int n_in,
                              void* d_out, int out_size, void* d_ws, size_t ws_size,
                              hipStream_t stream) {
    const float* x   = (const float*)d_in[0];
    const float* wih = (const float*)d_in[1];
    const float* whh = (const float*)d_in[2];
    const float* bih = (const float*)d_in[3];
    float* out = (float*)d_out;

    dim3 grid(B_, H_ / 256);   // 64 x 2 blocks, 8 waves each
    indrnn_fused_kernel<<<grid, 256, 0, stream>>>(x, wih, whh, bih, out);
}